// GUNet_24627342475436
// MI455X (gfx1250) — compile-verified
//
#include <hip/hip_runtime.h>
#include <hip/hip_bf16.h>

#define NN   3072
#define EE   49152
#define CIN  16
#define HIDD 64
#define COUTF 3
#define KP1  2000
#define KP2  1000
#define KP3  500

typedef __attribute__((ext_vector_type(2))) float v2f;
typedef __attribute__((ext_vector_type(8))) float v8f;

// ---------------- elementwise / small kernels ----------------

__global__ void k_zero(float* p, int n) {
  int i = blockIdx.x * blockDim.x + threadIdx.x;
  if (i < n) p[i] = 0.0f;
}

__global__ void k_copy(const float* __restrict__ s, float* __restrict__ d, int n) {
  int i = blockIdx.x * blockDim.x + threadIdx.x;
  if (i < n) d[i] = s[i];
}

__global__ void k_build_adj(float* __restrict__ A, const int* __restrict__ ei, int e, int n) {
  int i = blockIdx.x * blockDim.x + threadIdx.x;
  if (i < e) atomicAdd(&A[ei[i] * n + ei[e + i]], 1.0f);  // duplicate edges sum
}

__global__ void k_copy_diag1(const float* __restrict__ A, float* __restrict__ T, int n) {
  int i = blockIdx.x * blockDim.x + threadIdx.x;
  if (i < n * n) {
    int r = i / n, c = i - r * n;
    T[i] = (r == c) ? 1.0f : A[i];
  }
}

// LDS-tiled transpose D = S^T (n x n), 32x32 tiles, bank-conflict-free
__global__ void k_transpose(const float* __restrict__ S, float* __restrict__ D, int n) {
  __shared__ float tile[32][33];
  int bx = blockIdx.x * 32, by = blockIdx.y * 32;
  int tx = threadIdx.x, ty = threadIdx.y;
#pragma unroll
  for (int j = ty; j < 32; j += 8) {
    int r = by + j, c = bx + tx;
    tile[j][tx] = (r < n && c < n) ? S[r * n + c] : 0.0f;
  }
  __syncthreads();
#pragma unroll
  for (int j = ty; j < 32; j += 8) {
    int r = bx + j, c = by + tx;
    if (r < n && c < n) D[r * n + c] = tile[tx][j];
  }
}

// dis[r] = rsqrt(rowsum(A) + 2)   (improved GCN: A_hat = A + 2I)
__global__ void k_row_dis(const float* __restrict__ A, float* __restrict__ dis, int n) {
  __shared__ float red[256];
  int r = blockIdx.x;
  float s = 0.0f;
  for (int c = threadIdx.x; c < n; c += 256) s += A[r * n + c];
  red[threadIdx.x] = s;
  __syncthreads();
  for (int st = 128; st > 0; st >>= 1) {
    if (threadIdx.x < st) red[threadIdx.x] += red[threadIdx.x + st];
    __syncthreads();
  }
  if (threadIdx.x == 0) {
    float d = red[0] + 2.0f;
    dis[r] = (d > 0.0f) ? rsqrtf(d) : 0.0f;
  }
}

// O(n x cout) = X(n x cin) @ W(cin x cout), tiny K -> plain VALU
__global__ void k_mm_small(const float* __restrict__ X, const float* __restrict__ W,
                           float* __restrict__ O, int n, int cin, int cout) {
  int i = blockIdx.x * blockDim.x + threadIdx.x;
  if (i < n * cout) {
    int r = i / cout, c = i - r * cout;
    float s = 0.0f;
    for (int k = 0; k < cin; ++k) s += X[r * cin + k] * W[k * cout + c];
    O[i] = s;
  }
}

// Ot (c x n, transposed) : Ot[col][r] = X[r][col] * dis[r]
__global__ void k_scale_rows_t(const float* __restrict__ X, const float* __restrict__ dis,
                               float* __restrict__ Ot, int n, int c) {
  int i = blockIdx.x * blockDim.x + threadIdx.x;
  if (i < n * c) {
    int col = i / n, r = i - col * n;
    Ot[i] = X[r * c + col] * dis[r];
  }
}

__global__ void k_pnorm(const float* __restrict__ p, float* __restrict__ out) {
  __shared__ float red[64];
  float v = p[threadIdx.x];
  red[threadIdx.x] = v * v;
  __syncthreads();
  for (int st = 32; st > 0; st >>= 1) {
    if (threadIdx.x < st) red[threadIdx.x] += red[threadIdx.x + st];
    __syncthreads();
  }
  if (threadIdx.x == 0) out[0] = rsqrtf(red[0]);   // 1/||p||
}

__global__ void k_score(const float* __restrict__ x, const float* __restrict__ p,
                        const float* __restrict__ pinv, float* __restrict__ s, int n) {
  int i = blockIdx.x * blockDim.x + threadIdx.x;
  if (i < n) {
    float a = 0.0f;
    for (int k = 0; k < HIDD; ++k) a += x[i * HIDD + k] * p[k];
    s[i] = tanhf(a * pinv[0]);
  }
}

// Exact top-k via rank counting (matches lax.top_k descending order, ties by index)
__global__ void k_topk(const float* __restrict__ s, int* __restrict__ perm, int n, int k) {
  int i = blockIdx.x * blockDim.x + threadIdx.x;
  if (i < n) {
    float si = s[i];
    int r = 0;
    for (int j = 0; j < n; ++j) {
      float sj = s[j];
      r += (sj > si) || (sj == si && j < i);
    }
    if (r < k) perm[r] = i;
  }
}

__global__ void k_gather_x(const float* __restrict__ x, const float* __restrict__ s,
                           const int* __restrict__ perm, float* __restrict__ xo, int k) {
  int i = blockIdx.x * blockDim.x + threadIdx.x;
  if (i < k * HIDD) {
    int r = i / HIDD, c = i - r * HIDD;
    int pr = perm[r];
    xo[i] = x[pr * HIDD + c] * s[pr];
  }
}

// unpool skip: h[perm[r], :] += xs[r, :]   (perm entries unique)
__global__ void k_scatter_add(const float* __restrict__ xs, const int* __restrict__ perm,
                              float* __restrict__ h, int k) {
  int i = blockIdx.x * blockDim.x + threadIdx.x;
  if (i < k * HIDD) {
    int r = i / HIDD, c = i - r * HIDD;
    h[perm[r] * HIDD + c] += xs[i];
  }
}

__global__ void k_softmax3(const float* __restrict__ X, float* __restrict__ O, int n) {
  int i = blockIdx.x * blockDim.x + threadIdx.x;
  if (i < n) {
    float a = X[i * 3], b = X[i * 3 + 1], c = X[i * 3 + 2];
    float m = fmaxf(a, fmaxf(b, c));
    float ea = expf(a - m), eb = expf(b - m), ec = expf(c - m);
    float inv = 1.0f / (ea + eb + ec);
    O[i * 3] = ea * inv; O[i * 3 + 1] = eb * inv; O[i * 3 + 2] = ec * inv;
  }
}

// ---------------- fp32 WMMA GEMM (V_WMMA_F32_16X16X4_F32) ----------------
// C = A @ Bt^T.  A row-major (M x K, lda), Bt is B TRANSPOSED (Nc x K, ldbt)
// so both operand fragments are single 8-byte loads. One wave32 computes a
// 64x64 C macro-tile (4x4 fragments, 128 acc VGPRs). Branch-free K-loop:
// out-of-range row/col gather indices are CLAMPED (their products only land
// in C entries that the guarded store discards), and fragment loads are
// double-buffered so they hide behind the 16-WMMA block.
// Optional row/col gather = fused TopKPooling permutation. Epilogues:
//   mode 0: C = A@B
//   mode 1: C = A@B, zero diagonal          (augment: strip self loops of A^2)
//   mode 2: C = dis[r]*(A@B + 2*B[r,c]) + bias[c]        (GCN, improved)
//   mode 3: mode 2 followed by ReLU
__global__ __launch_bounds__(32)
void k_wmma_gemm(const float* __restrict__ A, int lda,
                 const float* __restrict__ Bt, int ldbt,
                 float* __restrict__ C, int ldc,
                 int M, int Nc, int K,
                 const int* __restrict__ rowperm, const int* __restrict__ colperm,
                 int mode, const float* __restrict__ dis, const float* __restrict__ bias) {
  const int lane = threadIdx.x;
  const int hi   = lane >> 4;    // lanes 16-31 hold K+2,K+3
  const int l16  = lane & 15;
  const int m0 = blockIdx.y * 64;
  const int n0 = blockIdx.x * 64;

  // per-lane K-contiguous base pointers (gather folded in, indices clamped)
  const float* pa[4];
  const float* pb[4];
#pragma unroll
  for (int t = 0; t < 4; ++t) {
    int r = m0 + t * 16 + l16;
    if (r >= M) r = M - 1;                       // clamp: result rows discarded
    int gr = rowperm ? rowperm[r] : r;
    pa[t] = A + gr * lda + hi * 2;
    int c = n0 + t * 16 + l16;
    if (c >= Nc) c = Nc - 1;                     // clamp: result cols discarded
    int gc = colperm ? colperm[c] : c;
    pb[t] = Bt + gc * ldbt + hi * 2;
  }

  v8f acc[4][4] = {};
  v2f af[4], bf[4];
#pragma unroll
  for (int t = 0; t < 4; ++t) {                  // prologue: K-step 0
    af[t] = *(const v2f*)pa[t];
    bf[t] = *(const v2f*)pb[t];
  }

  for (int k = 4; k < K; k += 4) {               // K always a multiple of 4 here
    v2f afn[4], bfn[4];
#pragma unroll
    for (int t = 0; t < 4; ++t) {                // issue next-step loads first
      afn[t] = *(const v2f*)(pa[t] + k);
      bfn[t] = *(const v2f*)(pb[t] + k);
    }
#pragma unroll
    for (int i = 0; i < 4; ++i)
#pragma unroll
      for (int j = 0; j < 4; ++j)
        acc[i][j] = __builtin_amdgcn_wmma_f32_16x16x4_f32(
            false, af[i], false, bf[j], (short)0, acc[i][j], false, false);
#pragma unroll
    for (int t = 0; t < 4; ++t) { af[t] = afn[t]; bf[t] = bfn[t]; }
  }
#pragma unroll
  for (int i = 0; i < 4; ++i)                    // drain last K-step
#pragma unroll
    for (int j = 0; j < 4; ++j)
      acc[i][j] = __builtin_amdgcn_wmma_f32_16x16x4_f32(
          false, af[i], false, bf[j], (short)0, acc[i][j], false, false);

#pragma unroll
  for (int i = 0; i < 4; ++i)
#pragma unroll
    for (int j = 0; j < 4; ++j)
#pragma unroll
      for (int v = 0; v < 8; ++v) {
        int r = m0 + i * 16 + hi * 8 + v;        // C: VGPR v = M=v / M=v+8 halves
        int c = n0 + j * 16 + l16;
        if (r < M && c < Nc) {
          float val = acc[i][j][v];
          if (mode >= 2) {
            val = dis[r] * (val + 2.0f * Bt[c * ldbt + r]) + bias[c];
            if (mode == 3) val = fmaxf(val, 0.0f);
          } else if (mode == 1 && r == c) {
            val = 0.0f;
          }
          C[r * ldc + c] = val;
        }
      }
}

// ---------------- host orchestration ----------------

extern "C" void kernel_launch(void* const* d_in, const int* in_sizes, int n_in,
                              void* d_out, int out_size, void* d_ws, size_t ws_size,
                              hipStream_t stream) {
  (void)in_sizes; (void)n_in; (void)out_size; (void)ws_size;
  const float* x_in = (const float*)d_in[0];
  const int*   ei   = (const int*)d_in[1];
  const float* Wd0 = (const float*)d_in[2];  const float* bd0 = (const float*)d_in[3];
  const float* Wd1 = (const float*)d_in[4];  const float* bd1 = (const float*)d_in[5];
  const float* Wd2 = (const float*)d_in[6];  const float* bd2 = (const float*)d_in[7];
  const float* Wd3 = (const float*)d_in[8];  const float* bd3 = (const float*)d_in[9];
  const float* p1  = (const float*)d_in[10];
  const float* p2  = (const float*)d_in[11];
  const float* p3  = (const float*)d_in[12];
  const float* Wu0 = (const float*)d_in[13]; const float* bu0 = (const float*)d_in[14];
  const float* Wu1 = (const float*)d_in[15]; const float* bu1 = (const float*)d_in[16];
  const float* Wu2 = (const float*)d_in[17]; const float* bu2 = (const float*)d_in[18];

  float* ws = (float*)d_ws;
  size_t off = 0;
  auto alloc = [&](size_t n) { float* p = ws + off; off += (n + 63) & ~(size_t)63; return p; };
  float* A0  = alloc((size_t)NN * NN);     // adjs[0]  (kept for up path)
  float* T   = alloc((size_t)NN * NN);     // diag-1 copy for augment (reused per level)
  float* Tt  = alloc((size_t)NN * NN);     // its transpose (B-side operand)
  float* A1  = alloc((size_t)KP1 * KP1);   // adjs[1]
  float* A2  = alloc((size_t)KP2 * KP2);   // adjs[2]
  float* A3  = alloc((size_t)KP3 * KP3);   // bottom adjacency
  float* xs0 = alloc(NN  * HIDD);
  float* xs1 = alloc(KP1 * HIDD);
  float* xs2 = alloc(KP2 * HIDD);
  float* xp  = alloc(NN  * HIDD);          // pooled-x scratch
  float* x3  = alloc(KP3 * HIDD);
  float* hb  = alloc(NN  * HIDD);          // res + unpool
  float* u   = alloc(NN  * HIDD);          // up-path gcn output
  float* xw  = alloc(NN  * HIDD);          // X@W scratch
  float* bmT = alloc(NN  * HIDD);          // dis-scaled XW, TRANSPOSED (Nf x n)
  float* logits = alloc(NN * COUTF);
  float* dis = alloc(NN);
  float* sc  = alloc(NN);
  float* pinv = alloc(64);
  int* perm1 = (int*)alloc(KP1);
  int* perm2 = (int*)alloc(KP2);
  int* perm3 = (int*)alloc(KP3);

  dim3 B(256);
  auto G  = [](int n) { return dim3((unsigned)((n + 255) / 256)); };
  auto GT = [](int nc, int m) { return dim3((unsigned)((nc + 63) / 64), (unsigned)((m + 63) / 64)); };
  auto GX = [](int n) { return dim3((unsigned)((n + 31) / 32), (unsigned)((n + 31) / 32)); };
  dim3 BX(32, 8);

  // dense adjacency from edge_index (scatter-add, duplicates sum)
  k_zero<<<G(NN * NN), B, 0, stream>>>(A0, NN * NN);
  k_build_adj<<<G(EE), B, 0, stream>>>(A0, ei, EE, NN);

  // ---- gcn0: relu(gcn(x, A0, Wd0)) -> xs0 ----
  k_row_dis<<<NN, 256, 0, stream>>>(A0, dis, NN);
  k_mm_small<<<G(NN * HIDD), B, 0, stream>>>(x_in, Wd0, xw, NN, CIN, HIDD);
  k_scale_rows_t<<<G(NN * HIDD), B, 0, stream>>>(xw, dis, bmT, NN, HIDD);
  k_wmma_gemm<<<GT(HIDD, NN), 32, 0, stream>>>(A0, NN, bmT, NN, xs0, HIDD,
      NN, HIDD, NN, nullptr, nullptr, 3, dis, bd0);

  // ---- down level 1: 3072 -> 2000 ----
  k_pnorm<<<1, 64, 0, stream>>>(p1, pinv);
  k_score<<<G(NN), B, 0, stream>>>(xs0, p1, pinv, sc, NN);
  k_topk<<<G(NN), B, 0, stream>>>(sc, perm1, NN, KP1);
  k_copy_diag1<<<G(NN * NN), B, 0, stream>>>(A0, T, NN);
  k_transpose<<<GX(NN), BX, 0, stream>>>(T, Tt, NN);
  // fused augment + pool: A1 = (T@T)[perm1][:,perm1], diag 0
  k_wmma_gemm<<<GT(KP1, KP1), 32, 0, stream>>>(T, NN, Tt, NN, A1, KP1,
      KP1, KP1, NN, perm1, perm1, 1, nullptr, nullptr);
  k_gather_x<<<G(KP1 * HIDD), B, 0, stream>>>(xs0, sc, perm1, xp, KP1);
  k_row_dis<<<KP1, 256, 0, stream>>>(A1, dis, KP1);
  k_mm_small<<<G(KP1 * HIDD), B, 0, stream>>>(xp, Wd1, xw, KP1, HIDD, HIDD);
  k_scale_rows_t<<<G(KP1 * HIDD), B, 0, stream>>>(xw, dis, bmT, KP1, HIDD);
  k_wmma_gemm<<<GT(HIDD, KP1), 32, 0, stream>>>(A1, KP1, bmT, KP1, xs1, HIDD,
      KP1, HIDD, KP1, nullptr, nullptr, 3, dis, bd1);

  // ---- down level 2: 2000 -> 1000 ----
  k_pnorm<<<1, 64, 0, stream>>>(p2, pinv);
  k_score<<<G(KP1), B, 0, stream>>>(xs1, p2, pinv, sc, KP1);
  k_topk<<<G(KP1), B, 0, stream>>>(sc, perm2, KP1, KP2);
  k_copy_diag1<<<G(KP1 * KP1), B, 0, stream>>>(A1, T, KP1);
  k_transpose<<<GX(KP1), BX, 0, stream>>>(T, Tt, KP1);
  k_wmma_gemm<<<GT(KP2, KP2), 32, 0, stream>>>(T, KP1, Tt, KP1, A2, KP2,
      KP2, KP2, KP1, perm2, perm2, 1, nullptr, nullptr);
  k_gather_x<<<G(KP2 * HIDD), B, 0, stream>>>(xs1, sc, perm2, xp, KP2);
  k_row_dis<<<KP2, 256, 0, stream>>>(A2, dis, KP2);
  k_mm_small<<<G(KP2 * HIDD), B, 0, stream>>>(xp, Wd2, xw, KP2, HIDD, HIDD);
  k_scale_rows_t<<<G(KP2 * HIDD), B, 0, stream>>>(xw, dis, bmT, KP2, HIDD);
  k_wmma_gemm<<<GT(HIDD, KP2), 32, 0, stream>>>(A2, KP2, bmT, KP2, xs2, HIDD,
      KP2, HIDD, KP2, nullptr, nullptr, 3, dis, bd2);

  // ---- down level 3: 1000 -> 500 ----
  k_pnorm<<<1, 64, 0, stream>>>(p3, pinv);
  k_score<<<G(KP2), B, 0, stream>>>(xs2, p3, pinv, sc, KP2);
  k_topk<<<G(KP2), B, 0, stream>>>(sc, perm3, KP2, KP3);
  k_copy_diag1<<<G(KP2 * KP2), B, 0, stream>>>(A2, T, KP2);
  k_transpose<<<GX(KP2), BX, 0, stream>>>(T, Tt, KP2);
  k_wmma_gemm<<<GT(KP3, KP3), 32, 0, stream>>>(T, KP2, Tt, KP2, A3, KP3,
      KP3, KP3, KP2, perm3, perm3, 1, nullptr, nullptr);
  k_gather_x<<<G(KP3 * HIDD), B, 0, stream>>>(xs2, sc, perm3, xp, KP3);
  k_row_dis<<<KP3, 256, 0, stream>>>(A3, dis, KP3);
  k_mm_small<<<G(KP3 * HIDD), B, 0, stream>>>(xp, Wd3, xw, KP3, HIDD, HIDD);
  k_scale_rows_t<<<G(KP3 * HIDD), B, 0, stream>>>(xw, dis, bmT, KP3, HIDD);
  k_wmma_gemm<<<GT(HIDD, KP3), 32, 0, stream>>>(A3, KP3, bmT, KP3, x3, HIDD,
      KP3, HIDD, KP3, nullptr, nullptr, 3, dis, bd3);

  // ---- up 0 (at 1000): h = xs2 + unpool(x3); relu(gcn(h, A2, Wu0)) -> u ----
  k_copy<<<G(KP2 * HIDD), B, 0, stream>>>(xs2, hb, KP2 * HIDD);
  k_scatter_add<<<G(KP3 * HIDD), B, 0, stream>>>(x3, perm3, hb, KP3);
  k_row_dis<<<KP2, 256, 0, stream>>>(A2, dis, KP2);
  k_mm_small<<<G(KP2 * HIDD), B, 0, stream>>>(hb, Wu0, xw, KP2, HIDD, HIDD);
  k_scale_rows_t<<<G(KP2 * HIDD), B, 0, stream>>>(xw, dis, bmT, KP2, HIDD);
  k_wmma_gemm<<<GT(HIDD, KP2), 32, 0, stream>>>(A2, KP2, bmT, KP2, u, HIDD,
      KP2, HIDD, KP2, nullptr, nullptr, 3, dis, bu0);

  // ---- up 1 (at 2000) ----
  k_copy<<<G(KP1 * HIDD), B, 0, stream>>>(xs1, hb, KP1 * HIDD);
  k_scatter_add<<<G(KP2 * HIDD), B, 0, stream>>>(u, perm2, hb, KP2);
  k_row_dis<<<KP1, 256, 0, stream>>>(A1, dis, KP1);
  k_mm_small<<<G(KP1 * HIDD), B, 0, stream>>>(hb, Wu1, xw, KP1, HIDD, HIDD);
  k_scale_rows_t<<<G(KP1 * HIDD), B, 0, stream>>>(xw, dis, bmT, KP1, HIDD);
  k_wmma_gemm<<<GT(HIDD, KP1), 32, 0, stream>>>(A1, KP1, bmT, KP1, u, HIDD,
      KP1, HIDD, KP1, nullptr, nullptr, 3, dis, bu1);

  // ---- up 2 (at 3072, C_OUT=3, no relu) + softmax ----
  k_copy<<<G(NN * HIDD), B, 0, stream>>>(xs0, hb, NN * HIDD);
  k_scatter_add<<<G(KP1 * HIDD), B, 0, stream>>>(u, perm1, hb, KP1);
  k_row_dis<<<NN, 256, 0, stream>>>(A0, dis, NN);
  k_mm_small<<<G(NN * COUTF), B, 0, stream>>>(hb, Wu2, xw, NN, HIDD, COUTF);
  k_scale_rows_t<<<G(NN * COUTF), B, 0, stream>>>(xw, dis, bmT, NN, COUTF);
  k_wmma_gemm<<<GT(COUTF, NN), 32, 0, stream>>>(A0, NN, bmT, NN, logits, COUTF,
      NN, COUTF, NN, nullptr, nullptr, 2, dis, bu2);
  k_softmax3<<<G(NN), B, 0, stream>>>(logits, (float*)d_out, NN);
}